// get_model_83623013253531
// MI455X (gfx1250) — compile-verified
//
#include <hip/hip_runtime.h>
#include <hip/hip_bf16.h>

// ---------------------------------------------------------------------------
// Types for WMMA fragments (f32 path: A/B = 16x4 / 4x16 -> 2 VGPRs, C/D = 8)
// ---------------------------------------------------------------------------
typedef float v2f __attribute__((ext_vector_type(2)));
typedef float v8f __attribute__((ext_vector_type(8)));

#define H_IMG 64
#define W_IMG 64
#define NPIX  4096
#define DEPTH_RES 100

// ---------------------------------------------------------------------------
// Bicubic sampling (a = -0.75), matching the reference's clip+mask semantics:
// out-of-range taps contribute 0.
// ---------------------------------------------------------------------------
__device__ __forceinline__ void cubic_w(float t, float w[4]) {
    const float a = -0.75f;
    float t1 = t + 1.0f;
    float t2 = 2.0f - t;
    float s  = 1.0f - t;
    w[0] = ((a * t1 - 5.0f * a) * t1 + 8.0f * a) * t1 - 4.0f * a;
    w[1] = ((a + 2.0f) * t - (a + 3.0f)) * t * t + 1.0f;
    w[2] = ((a + 2.0f) * s - (a + 3.0f)) * s * s + 1.0f;
    w[3] = ((a * t2 - 5.0f * a) * t2 + 8.0f * a) * t2 - 4.0f * a;
}

__device__ __forceinline__ float bicubic64(const float* __restrict__ img,
                                           float sx, float sy) {
    float fx0 = floorf(sx);
    float fy0 = floorf(sy);
    float wx[4], wy[4];
    cubic_w(sx - fx0, wx);
    cubic_w(sy - fy0, wy);
    int ix0 = (int)fx0;
    int iy0 = (int)fy0;
    float acc = 0.0f;
#pragma unroll
    for (int j = 0; j < 4; ++j) {
        int py = iy0 + j - 1;
        if (py < 0 || py > 63) continue;     // masked -> contributes 0
        float row = 0.0f;
#pragma unroll
        for (int i = 0; i < 4; ++i) {
            int px = ix0 + i - 1;
            if (px < 0 || px > 63) continue; // masked -> contributes 0
            row += wx[i] * img[py * W_IMG + px];
        }
        acc += wy[j] * row;
    }
    return acc;
}

// ---------------------------------------------------------------------------
// Stage 1: grayscale per source view: gray = sum_c coeff[c]*lf[c] / 3
// lf layout: (1,3,2,2,64,64)
// ---------------------------------------------------------------------------
__global__ void gray_kernel(const float* __restrict__ lf, float* __restrict__ gray) {
    int id = blockIdx.x * blockDim.x + threadIdx.x;
    if (id >= 4 * NPIX) return;
    int s   = id >> 12;
    int pix = id & (NPIX - 1);
    float r = lf[0 * 4 * NPIX + s * NPIX + pix];
    float g = lf[1 * 4 * NPIX + s * NPIX + pix];
    float b = lf[2 * 4 * NPIX + s * NPIX + pix];
    gray[id] = (0.299f * r + 0.587f * g + 0.114f * b) * (1.0f / 3.0f);
}

// ---------------------------------------------------------------------------
// Stage 2: depth features (defocus + correspondence).
// feats layout: [t(4)][ch(200)][pix], ch 0..99 = defocus, 100..199 = corresp.
// ---------------------------------------------------------------------------
__global__ void depth_feat_kernel(const float* __restrict__ gray,
                                  float* __restrict__ feats) {
    int id = blockIdx.x * blockDim.x + threadIdx.x;
    if (id >= 4 * DEPTH_RES * NPIX) return;
    int pix = id & (NPIX - 1);
    int d   = (id >> 12) % DEPTH_RES;
    int t   = id / (DEPTH_RES * NPIX);
    int tu = t >> 1, tv = t & 1;
    int y = pix >> 6, x = pix & 63;

    float depth = -21.0f + (float)d * (42.0f / 99.0f);
    const float sc = 63.0f / 64.0f;   // (H-1)/H normalization factor

    float vals[4];
#pragma unroll
    for (int s = 0; s < 4; ++s) {
        int su = s >> 1, sv = s & 1;
        float cy = (float)y + depth * (float)(tu - su);
        float cx = (float)x + depth * (float)(tv - sv);
        vals[s] = bicubic64(gray + s * NPIX, cx * sc, cy * sc);
    }

    float ssum = 0.0f;
    int cnt = 0;
#pragma unroll
    for (int s = 0; s < 4; ++s) {
        ssum += vals[s];
        cnt += (vals[s] != 0.0f) ? 1 : 0;
    }
    float defocus = (cnt > 0) ? (ssum / (float)cnt) : 0.0f;
    float mu = ssum * 0.25f;
    float var = 0.0f;
#pragma unroll
    for (int s = 0; s < 4; ++s) {
        float dv = vals[s] - mu;
        var += dv * dv;
    }
    var *= (1.0f / 3.0f);
    float corresp = (cnt == 4) ? var : 0.0f;

    feats[(size_t)t * (200 * NPIX) + (size_t)d * NPIX + pix]               = defocus;
    feats[(size_t)t * (200 * NPIX) + (size_t)(DEPTH_RES + d) * NPIX + pix] = corresp;
}

// ---------------------------------------------------------------------------
// Implicit-GEMM convolution using V_WMMA_F32_16X16X4_F32 with cooperative
// LDS staging of the B (im2col) operand.
//
// Block = 128 threads = 4 waves. A block owns one 16-pixel tile (fixed y,
// 16 consecutive x) and a group of up to EIGHT oc-tiles (TWO per wave). K is
// marched in chunks of BK=64:
//   1) all 128 threads cooperatively build B_lds[64][16] (bounds-checked
//      im2col patch, zeros outside image / past Ktot) - amortized over 8
//      oc-tiles
//   2) each wave runs 16 WMMA k-steps; the B fragment is loaded once from
//      LDS (2x ds_load_b32) and reused by TWO WMMAs (one per oc-tile), with
//      A fetched as aligned float2 from the two weight rows (L2-resident,
//      next chunk prefetched via global_prefetch_b8)
//
// Fragment layouts (ISA 7.12.2, 32-bit):
//   A 16x4 : lanes 0-15 hold {K=k0,k0+1} in v0/v1; lanes 16-31 hold {k0+2,k0+3}
//   B 4x16 : mirrored (v0: K=k0 lanes 0-15 / K=k0+2 lanes 16-31; v1: +1)
//   C/D    : VGPR r -> M = r (lanes 0-15) / r+8 (lanes 16-31), N = lane%16
// ---------------------------------------------------------------------------
template <int KH, int KW>
__global__ __launch_bounds__(128) void conv_wmma_kernel(
    const float* __restrict__ in, const float* __restrict__ wgt,
    const float* __restrict__ bias, float* __restrict__ out,
    int nimg, int IC, int OC, int pad,
    long long in_img_stride, long long out_img_stride, long long out_oc_stride,
    int relu, float scale)
{
    constexpr int KK = KH * KW;
    constexpr int BK = 64;                    // K-chunk staged in LDS
    __shared__ float Blds[BK * 16];           // 4 KB, row stride 16 floats

    const int octiles  = (OC + 15) >> 4;
    const int ocgroups = (octiles + 7) >> 3;  // 8 oc-tiles per block
    const int pixtiles = NPIX / 16;           // 256

    int blk = blockIdx.x;
    const int img   = blk / (pixtiles * ocgroups);
    int rem         = blk - img * (pixtiles * ocgroups);
    const int ocg   = rem / pixtiles;
    const int ptile = rem - ocg * pixtiles;
    const int y  = ptile >> 2;                // 4 pixel-tiles per image row
    const int x0 = (ptile & 3) << 4;

    const int tid    = threadIdx.x;
    const int waveid = tid >> 5;
    const int lane   = tid & 31;
    const int octile0 = ocg * 8 + waveid * 2; // this wave's two oc-tiles
    const int octile1 = octile0 + 1;
    const bool wactive = (octile0 < octiles); // wave-uniform
    const bool tile1on = (octile1 < octiles); // wave-uniform

    const int Ktot = IC * KK;
    const int half = lane >> 4;               // 0: K pair (k,k+1), 1: (k+2,k+3)
    const int l16  = lane & 15;
    const int m0   = octile0 * 16 + l16;      // A rows (output channels)
    const int m1   = octile1 * 16 + l16;
    const bool mval0 = wactive && (m0 < OC);
    const bool mval1 = tile1on && (m1 < OC);
    const float* __restrict__ wrow0 = wgt + (long long)m0 * Ktot;
    const float* __restrict__ wrow1 = wgt + (long long)m1 * Ktot;
    const float* __restrict__ inb   = in + (long long)img * in_img_stride;

    v8f acc0 = {0.f, 0.f, 0.f, 0.f, 0.f, 0.f, 0.f, 0.f};
    v8f acc1 = {0.f, 0.f, 0.f, 0.f, 0.f, 0.f, 0.f, 0.f};

    for (int k0c = 0; k0c < Ktot; k0c += BK) {
        // ---- cooperative B fill: 1024 elements / 128 threads = 8 each ----
#pragma unroll
        for (int e = tid; e < BK * 16; e += 128) {
            const int kk = e >> 4;            // local k
            const int n  = e & 15;            // pixel column within tile
            const int k  = k0c + kk;
            float v = 0.0f;
            if (k < Ktot) {
                const int ic = k / KK;
                const int r  = k - ic * KK;
                const int ky = r / KW;
                const int kx = r - ky * KW;
                const int yy = y + ky - pad;
                const int xx = x0 + n + kx - pad;
                if ((unsigned)yy < (unsigned)H_IMG && (unsigned)xx < (unsigned)W_IMG)
                    v = inb[(long long)ic * NPIX + yy * W_IMG + xx];
            }
            Blds[e] = v;
        }
        __syncthreads();

        // ---- 16 WMMA k-steps over the staged chunk ----
        if (wactive) {
            // prefetch next K-chunk of weights (global_prefetch_b8)
            if (mval0 && (k0c + BK) < Ktot)
                __builtin_prefetch(wrow0 + k0c + BK, 0, 3);
            if (mval1 && (k0c + BK) < Ktot)
                __builtin_prefetch(wrow1 + k0c + BK, 0, 3);

#pragma unroll
            for (int ks = 0; ks < BK / 4; ++ks) {
                const int klocal = ks * 4 + 2 * half; // this lane's first K
                const int ka     = k0c + klocal;

                // B fragment: loaded once, reused by both WMMAs
                v2f bfrag;
                bfrag.x = Blds[klocal * 16 + l16];
                bfrag.y = Blds[(klocal + 1) * 16 + l16];

                v2f afrag0;
                if (mval0 && (ka + 1) < Ktot) {
                    afrag0 = *reinterpret_cast<const v2f*>(wrow0 + ka); // 8B aligned
                } else {
                    afrag0.x = (mval0 && ka < Ktot) ? wrow0[ka] : 0.0f;
                    afrag0.y = 0.0f;
                }
                acc0 = __builtin_amdgcn_wmma_f32_16x16x4_f32(
                    false, afrag0, false, bfrag, (short)0, acc0, false, false);

                if (tile1on) {
                    v2f afrag1;
                    if (mval1 && (ka + 1) < Ktot) {
                        afrag1 = *reinterpret_cast<const v2f*>(wrow1 + ka);
                    } else {
                        afrag1.x = (mval1 && ka < Ktot) ? wrow1[ka] : 0.0f;
                        afrag1.y = 0.0f;
                    }
                    acc1 = __builtin_amdgcn_wmma_f32_16x16x4_f32(
                        false, afrag1, false, bfrag, (short)0, acc1, false, false);
                }
            }
        }
        __syncthreads();
    }

    // ---- epilogue: bias, optional relu, optional scale, strided store ----
    if (wactive) {
#pragma unroll
        for (int r = 0; r < 8; ++r) {
            const int M   = r + 8 * half;
            const int oc0 = octile0 * 16 + M;
            if (oc0 < OC) {
                float v = acc0[r] + bias[oc0];
                if (relu) v = fmaxf(v, 0.0f);
                v *= scale;
                out[(long long)img * out_img_stride +
                    (long long)oc0 * out_oc_stride + y * W_IMG + x0 + l16] = v;
            }
            const int oc1 = octile1 * 16 + M;
            if (tile1on && oc1 < OC) {
                float v = acc1[r] + bias[oc1];
                if (relu) v = fmaxf(v, 0.0f);
                v *= scale;
                out[(long long)img * out_img_stride +
                    (long long)oc1 * out_oc_stride + y * W_IMG + x0 + l16] = v;
            }
        }
    }
}

// ---------------------------------------------------------------------------
// Stage 4: warp the 4 input views (3 channels each) to 49 target views using
// the predicted per-source-view disparity.
// warped layout: [t49][ch12][pix], t49 = t1*7+t2, ch = (su*2+sv)*3 + c
// ---------------------------------------------------------------------------
__global__ void warp_kernel(const float* __restrict__ lf,
                            const float* __restrict__ disp,
                            float* __restrict__ warped) {
    int id = blockIdx.x * blockDim.x + threadIdx.x;
    if (id >= 49 * 12 * NPIX) return;
    int pix = id & (NPIX - 1);
    int ch  = (id >> 12) % 12;
    int t   = id / (12 * NPIX);
    int t1 = t / 7, t2 = t - t1 * 7;
    int sview = ch / 3;
    int c     = ch - sview * 3;
    int su = sview >> 1, sv = sview & 1;
    int y = pix >> 6, x = pix & 63;

    float d  = disp[sview * NPIX + pix];
    float dU = (float)t1 - 6.0f * (float)su;   // af = (7-1)/(2-1) = 6
    float dV = (float)t2 - 6.0f * (float)sv;
    float cy = (float)y - dU * d;
    float cx = (float)x - dV * d;
    const float sc = 63.0f / 64.0f;

    warped[id] = bicubic64(lf + (size_t)c * 4 * NPIX + (size_t)sview * NPIX,
                           cx * sc, cy * sc);
}

// ---------------------------------------------------------------------------
// Host-side conv launcher
// ---------------------------------------------------------------------------
template <int KH, int KW>
static void run_conv(const float* in, const float* wgt, const float* bias,
                     float* out, int nimg, int IC, int OC, int pad,
                     long long in_is, long long out_is, long long out_os,
                     int relu, float scale, hipStream_t stream) {
    int octiles  = (OC + 15) / 16;
    int ocgroups = (octiles + 7) / 8;
    long long blocks = (long long)nimg * ocgroups * (NPIX / 16);
    conv_wmma_kernel<KH, KW><<<(int)blocks, 128, 0, stream>>>(
        in, wgt, bias, out, nimg, IC, OC, pad, in_is, out_is, out_os, relu, scale);
}

// ---------------------------------------------------------------------------
// kernel_launch
// Input order: lf, dw1,db1, dw2,db2, dw3,db3, dw4,db4,
//              cw1,cb1, cw2,cb2, cw3,cb3, cw4,cb4
// ---------------------------------------------------------------------------
extern "C" void kernel_launch(void* const* d_in, const int* in_sizes, int n_in,
                              void* d_out, int out_size, void* d_ws, size_t ws_size,
                              hipStream_t stream) {
    (void)in_sizes; (void)n_in; (void)out_size; (void)ws_size;

    const float* lf  = (const float*)d_in[0];
    const float* dw1 = (const float*)d_in[1];  const float* db1 = (const float*)d_in[2];
    const float* dw2 = (const float*)d_in[3];  const float* db2 = (const float*)d_in[4];
    const float* dw3 = (const float*)d_in[5];  const float* db3 = (const float*)d_in[6];
    const float* dw4 = (const float*)d_in[7];  const float* db4 = (const float*)d_in[8];
    const float* cw1 = (const float*)d_in[9];  const float* cb1 = (const float*)d_in[10];
    const float* cw2 = (const float*)d_in[11]; const float* cb2 = (const float*)d_in[12];
    const float* cw3 = (const float*)d_in[13]; const float* cb3 = (const float*)d_in[14];
    const float* cw4 = (const float*)d_in[15]; const float* cb4 = (const float*)d_in[16];
    float* outp = (float*)d_out;

    // Workspace arenas (floats). Lifetime-based aliasing:
    //   arenaA: feats (4x200x4096)   -> later cact1 (49x100x4096)
    //   arenaB: dact1 (4x100x4096)   -> later cact2 (49x100x4096)
    //   arenaC: dact2 (4x100x4096)   -> later cact3 (49x50x4096)
    float* ws = (float*)d_ws;
    size_t off = 0;
    float* gray   = ws + off; off += (size_t)4 * NPIX;              // 16 K
    float* disp   = ws + off; off += (size_t)4 * NPIX;              // 16 K
    float* dact3  = ws + off; off += (size_t)4 * 50 * NPIX;         // 0.8 M
    float* warped = ws + off; off += (size_t)49 * 12 * NPIX;        // 2.4 M
    float* arenaA = ws + off; off += (size_t)49 * 100 * NPIX;       // 20.1 M
    float* arenaB = ws + off; off += (size_t)49 * 100 * NPIX;       // 20.1 M
    float* arenaC = ws + off; off += (size_t)49 * 50  * NPIX;       // 10.0 M

    float* feats = arenaA;
    float* dact1 = arenaB;
    float* dact2 = arenaC;
    float* cact1 = arenaA;
    float* cact2 = arenaB;
    float* cact3 = arenaC;

    // 1) grayscale views
    gray_kernel<<<(4 * NPIX + 255) / 256, 256, 0, stream>>>(lf, gray);

    // 2) depth features (defocus + correspondence), 1.64M threads
    depth_feat_kernel<<<(4 * DEPTH_RES * NPIX + 255) / 256, 256, 0, stream>>>(gray, feats);

    // 3) depth CNN (WMMA implicit GEMM, LDS-staged B, 2 oc-tiles/wave)
    run_conv<7, 7>(feats, dw1, db1, dact1, 4, 200, 100, 3,
                   (long long)200 * NPIX, (long long)100 * NPIX, NPIX, 1, 1.0f, stream);
    run_conv<5, 5>(dact1, dw2, db2, dact2, 4, 100, 100, 2,
                   (long long)100 * NPIX, (long long)100 * NPIX, NPIX, 1, 1.0f, stream);
    run_conv<3, 3>(dact2, dw3, db3, dact3, 4, 100, 50, 1,
                   (long long)100 * NPIX, (long long)50 * NPIX, NPIX, 1, 1.0f, stream);
    // 1x1 -> disparity, scaled by 1/(ANG_OUT-1)
    run_conv<1, 1>(dact3, dw4, db4, disp, 4, 50, 1, 0,
                   (long long)50 * NPIX, NPIX, NPIX, 0, 1.0f / 6.0f, stream);

    // 4) warp 4 views x 3 channels to 49 target views
    warp_kernel<<<(49 * 12 * NPIX + 255) / 256, 256, 0, stream>>>(lf, disp, warped);

    // 5) color CNN (WMMA implicit GEMM, LDS-staged B, 2 oc-tiles/wave)
    run_conv<7, 7>(warped, cw1, cb1, cact1, 49, 12, 100, 3,
                   (long long)12 * NPIX, (long long)100 * NPIX, NPIX, 1, 1.0f, stream);
    run_conv<5, 5>(cact1, cw2, cb2, cact2, 49, 100, 100, 2,
                   (long long)100 * NPIX, (long long)100 * NPIX, NPIX, 1, 1.0f, stream);
    run_conv<3, 3>(cact2, cw3, cb3, cact3, 49, 100, 50, 1,
                   (long long)100 * NPIX, (long long)50 * NPIX, NPIX, 1, 1.0f, stream);
    // final 1x1 -> d_out in (3, 7, 7, 64, 64) order:
    //   out[c][t1][t2][y][x]  => img(t49) stride = 4096, oc(c) stride = 49*4096
    run_conv<1, 1>(cact3, cw4, cb4, outp, 49, 50, 3, 0,
                   (long long)50 * NPIX, NPIX, (long long)49 * NPIX, 0, 1.0f, stream);
}